// FusedThorMoE_52304111730968
// MI455X (gfx1250) — compile-verified
//
#include <hip/hip_runtime.h>
#include <hip/hip_bf16.h>
#include <math.h>

// ---------------------------------------------------------------------------
// Fused MoE block for MI455X (gfx1250, wave32).
//  - fp32 matrix math on V_WMMA_F32_16X16X4_F32 (native CDNA5 fp32 path)
//  - double-buffered K-loop staged with GLOBAL_LOAD_ASYNC_TO_LDS_B128,
//    synchronized with s_wait_asynccnt (ASYNCcnt), sync fallback if the
//    builtins are unavailable.
// ---------------------------------------------------------------------------

#define E_  16
#define H_  512
#define I_  1024
#define N_  16384          // B*S
#define C_  3072           // per-expert capacity (3x mean load)

#define TM 128             // block tile M
#define TN 128             // block tile N
#define KC 16              // K chunk staged in LDS
#define KA (KC + 4)        // A row pitch: 20 floats = 80B (16B multiple)

typedef float v2f __attribute__((ext_vector_type(2)));
typedef float v8f __attribute__((ext_vector_type(8)));
typedef int   v4i __attribute__((ext_vector_type(4)));

#if __has_builtin(__builtin_amdgcn_global_load_async_to_lds_b128) && \
    __has_builtin(__builtin_amdgcn_s_wait_asynccnt)
#define USE_ASYNC 1
#else
#define USE_ASYNC 0
#endif

#if USE_ASYNC
typedef __attribute__((address_space(1))) v4i GV4;   // global v4i
typedef __attribute__((address_space(3))) v4i LV4;   // LDS v4i
#endif

// 16-byte global -> LDS copy (async on CDNA5)
static __device__ __forceinline__ void cp16(const float* g, float* l) {
#if USE_ASYNC
    __builtin_amdgcn_global_load_async_to_lds_b128((GV4*)g, (LV4*)l, 0, 0);
#else
    *(float4*)l = *(const float4*)g;
#endif
}
static __device__ __forceinline__ void wait_async_le4() {
#if USE_ASYNC
    __builtin_amdgcn_s_wait_asynccnt(4);   // previous chunk's 4 copies done
#endif
}
static __device__ __forceinline__ void wait_async_le0() {
#if USE_ASYNC
    __builtin_amdgcn_s_wait_asynccnt(0);
#endif
}

static __device__ __forceinline__ v8f wmma_f32(v2f a, v2f b, v8f c) {
    // 8 args: (neg_a, A, neg_b, B, c_mod, C, reuse_a, reuse_b)
    return __builtin_amdgcn_wmma_f32_16x16x4_f32(false, a, false, b,
                                                 (short)0, c, false, false);
}

// ---------------------------------------------------------------------------
// 1) init: zero expert counters, fill slot->token map with -1
// ---------------------------------------------------------------------------
__global__ __launch_bounds__(256) void moe_init(int* __restrict__ counters,
                                                int* __restrict__ tos) {
    int idx = blockIdx.x * 256 + threadIdx.x;
    if (idx < E_ * C_) tos[idx] = -1;
    if (idx < E_) counters[idx] = 0;
}

// ---------------------------------------------------------------------------
// 2) route: injective token->slot assignment via atomics (order-free: the
//    scatter/gather round-trip makes in-queue position irrelevant).
// ---------------------------------------------------------------------------
__global__ __launch_bounds__(256) void moe_route(const int* __restrict__ assign,
                                                 int* __restrict__ counters,
                                                 int* __restrict__ tos) {
    int n = blockIdx.x * 256 + threadIdx.x;
    if (n >= N_) return;
    int e = assign[n];
    int p = atomicAdd(&counters[e], 1);
    if (p < C_) tos[e * C_ + p] = n;
}

// ---------------------------------------------------------------------------
// 3) seed out = hidden_states (handles dropped tokens: residual-only -> LN(x))
// ---------------------------------------------------------------------------
__global__ __launch_bounds__(256) void moe_seed_out(const float* __restrict__ x,
                                                    float* __restrict__ out) {
    const int total4 = (N_ * H_) / 4;
    for (int i = blockIdx.x * 256 + threadIdx.x; i < total4; i += gridDim.x * 256) {
        reinterpret_cast<float4*>(out)[i] = reinterpret_cast<const float4*>(x)[i];
    }
}

// ---------------------------------------------------------------------------
// 4) GEMM1: h[e] = gelu( gather(x)[C_,H_] @ W1[e][H_,I_] + b1[e] )
//    grid = (I_/TN, C_/TM, E_), 256 threads (8 waves, 2x4), wave tile 64x32.
//    Rows without a token load garbage (clamped addr) -> only pollute their
//    own output row, which is never gathered back. Keeps staging fully async.
// ---------------------------------------------------------------------------
__global__ __launch_bounds__(256) void moe_gemm1(const float* __restrict__ x,
                                                 const float* __restrict__ W1,
                                                 const float* __restrict__ b1,
                                                 const int*   __restrict__ tos,
                                                 const int*   __restrict__ counters,
                                                 float*       __restrict__ h) {
    const int e  = blockIdx.z;
    const int m0 = blockIdx.y * TM;
    const int n0 = blockIdx.x * TN;
    int cnt = counters[e]; if (cnt > C_) cnt = C_;
    if (m0 >= cnt) return;                       // skip empty capacity tiles

    __shared__ float As[2][TM][KA];              // [m][k], pitch 20
    __shared__ float Bs[2][KC][TN + 4];          // [k][n]

    const int tid  = threadIdx.x;
    const int lane = tid & 31;
    const int wave = tid >> 5;
    const int wm   = (wave >> 2) * 64;           // 0 / 64
    const int wn   = (wave & 3) * 32;            // 0..96
    const int half = lane >> 4;                  // ISA lane-half
    const int l16  = lane & 15;

    // per-thread staging slices (fixed across chunks)
    const int am  = tid >> 1;                    // A row
    const int ak8 = (tid & 1) * 8;               // A k-subchunk
    int tok = tos[e * C_ + m0 + am]; if (tok < 0) tok = 0;
    const float* aBase = x + (size_t)tok * H_ + ak8;
    const int bk  = tid >> 4;                    // B k-row
    const int bn8 = (tid & 15) * 8;              // B n-subchunk
    const float* bBase = W1 + (size_t)e * H_ * I_ + (size_t)bk * I_ + n0 + bn8;

    v8f acc[4][2];
    #pragma unroll
    for (int mi = 0; mi < 4; ++mi)
        #pragma unroll
        for (int ni = 0; ni < 2; ++ni) acc[mi][ni] = (v8f)0.0f;

    // prologue: stage chunk 0 into buffer 0
    cp16(aBase,     &As[0][am][ak8]);
    cp16(aBase + 4, &As[0][am][ak8 + 4]);
    cp16(bBase,     &Bs[0][bk][bn8]);
    cp16(bBase + 4, &Bs[0][bk][bn8 + 4]);

    const int NCHUNK = H_ / KC;                  // 32
    for (int c = 0; c < NCHUNK; ++c) {
        const int cur = c & 1;
        if (c + 1 < NCHUNK) {                    // issue next chunk first
            const int nb = cur ^ 1;
            const int k1 = (c + 1) * KC;
            cp16(aBase + k1,                  &As[nb][am][ak8]);
            cp16(aBase + k1 + 4,              &As[nb][am][ak8 + 4]);
            cp16(bBase + (size_t)k1 * I_,     &Bs[nb][bk][bn8]);
            cp16(bBase + (size_t)k1 * I_ + 4, &Bs[nb][bk][bn8 + 4]);
            wait_async_le4();                    // chunk c complete (in-order)
        } else {
            wait_async_le0();
        }
        __syncthreads();

        #pragma unroll
        for (int kk = 0; kk < KC; kk += 4) {
            v2f a[4], b[2];
            #pragma unroll
            for (int mi = 0; mi < 4; ++mi)       // single aligned ds_load_b64
                a[mi] = *(const v2f*)&As[cur][wm + mi * 16 + l16][kk + 2 * half];
            #pragma unroll
            for (int ni = 0; ni < 2; ++ni) {
                b[ni].x = Bs[cur][kk + 2 * half + 0][wn + ni * 16 + l16];
                b[ni].y = Bs[cur][kk + 2 * half + 1][wn + ni * 16 + l16];
            }
            #pragma unroll
            for (int mi = 0; mi < 4; ++mi)
                #pragma unroll
                for (int ni = 0; ni < 2; ++ni)
                    acc[mi][ni] = wmma_f32(a[mi], b[ni], acc[mi][ni]);
        }
        __syncthreads();
    }

    // ---- epilogue: bias + exact GeLU, store h ----
    #pragma unroll
    for (int ni = 0; ni < 2; ++ni) {
        const int n = n0 + wn + ni * 16 + l16;
        const float bias = b1[(size_t)e * I_ + n];
        #pragma unroll
        for (int mi = 0; mi < 4; ++mi) {
            #pragma unroll
            for (int r = 0; r < 8; ++r) {
                const int m = m0 + wm + mi * 16 + half * 8 + r;  // C/D layout
                float v = acc[mi][ni][r] + bias;
                v = 0.5f * v * (1.0f + erff(v * 0.70710678118654752f));
                h[((size_t)e * C_ + m) * I_ + n] = v;
            }
        }
    }
}

// ---------------------------------------------------------------------------
// 5) GEMM2: out[token] = h[e] @ W2[e] + b2[e] + x[token]  (scatter per token)
//    grid = (H_/TN, C_/TM, E_)
// ---------------------------------------------------------------------------
__global__ __launch_bounds__(256) void moe_gemm2(const float* __restrict__ h,
                                                 const float* __restrict__ W2,
                                                 const float* __restrict__ b2,
                                                 const float* __restrict__ x,
                                                 const int*   __restrict__ tos,
                                                 const int*   __restrict__ counters,
                                                 float*       __restrict__ out) {
    const int e  = blockIdx.z;
    const int m0 = blockIdx.y * TM;
    const int n0 = blockIdx.x * TN;
    int cnt = counters[e]; if (cnt > C_) cnt = C_;
    if (m0 >= cnt) return;

    __shared__ float As[2][TM][KA];
    __shared__ float Bs[2][KC][TN + 4];
    __shared__ int   rowTok[TM];

    const int tid  = threadIdx.x;
    if (tid < TM) rowTok[tid] = tos[e * C_ + m0 + tid];

    const int lane = tid & 31;
    const int wave = tid >> 5;
    const int wm   = (wave >> 2) * 64;
    const int wn   = (wave & 3) * 32;
    const int half = lane >> 4;
    const int l16  = lane & 15;

    const int am  = tid >> 1;
    const int ak8 = (tid & 1) * 8;
    const float* aBase = h + ((size_t)e * C_ + m0 + am) * I_ + ak8;
    const int bk  = tid >> 4;
    const int bn8 = (tid & 15) * 8;
    const float* bBase = W2 + (size_t)e * I_ * H_ + (size_t)bk * H_ + n0 + bn8;

    v8f acc[4][2];
    #pragma unroll
    for (int mi = 0; mi < 4; ++mi)
        #pragma unroll
        for (int ni = 0; ni < 2; ++ni) acc[mi][ni] = (v8f)0.0f;

    cp16(aBase,     &As[0][am][ak8]);
    cp16(aBase + 4, &As[0][am][ak8 + 4]);
    cp16(bBase,     &Bs[0][bk][bn8]);
    cp16(bBase + 4, &Bs[0][bk][bn8 + 4]);

    const int NCHUNK = I_ / KC;                  // 64
    for (int c = 0; c < NCHUNK; ++c) {
        const int cur = c & 1;
        if (c + 1 < NCHUNK) {
            const int nb = cur ^ 1;
            const int k1 = (c + 1) * KC;
            cp16(aBase + k1,                  &As[nb][am][ak8]);
            cp16(aBase + k1 + 4,              &As[nb][am][ak8 + 4]);
            cp16(bBase + (size_t)k1 * H_,     &Bs[nb][bk][bn8]);
            cp16(bBase + (size_t)k1 * H_ + 4, &Bs[nb][bk][bn8 + 4]);
            wait_async_le4();
        } else {
            wait_async_le0();
        }
        __syncthreads();

        #pragma unroll
        for (int kk = 0; kk < KC; kk += 4) {
            v2f a[4], b[2];
            #pragma unroll
            for (int mi = 0; mi < 4; ++mi)
                a[mi] = *(const v2f*)&As[cur][wm + mi * 16 + l16][kk + 2 * half];
            #pragma unroll
            for (int ni = 0; ni < 2; ++ni) {
                b[ni].x = Bs[cur][kk + 2 * half + 0][wn + ni * 16 + l16];
                b[ni].y = Bs[cur][kk + 2 * half + 1][wn + ni * 16 + l16];
            }
            #pragma unroll
            for (int mi = 0; mi < 4; ++mi)
                #pragma unroll
                for (int ni = 0; ni < 2; ++ni)
                    acc[mi][ni] = wmma_f32(a[mi], b[ni], acc[mi][ni]);
        }
        __syncthreads();
    }

    // ---- epilogue: bias + residual, scatter straight to per-token output ----
    #pragma unroll
    for (int ni = 0; ni < 2; ++ni) {
        const int n = n0 + wn + ni * 16 + l16;
        const float bias = b2[(size_t)e * H_ + n];
        #pragma unroll
        for (int mi = 0; mi < 4; ++mi) {
            #pragma unroll
            for (int r = 0; r < 8; ++r) {
                const int m = wm + mi * 16 + half * 8 + r;
                const int t = rowTok[m];
                if (t >= 0) {
                    const size_t o = (size_t)t * H_ + n;
                    out[o] = acc[mi][ni][r] + bias + x[o];
                }
            }
        }
    }
}

// ---------------------------------------------------------------------------
// 6) in-place LayerNorm over each token row of out
// ---------------------------------------------------------------------------
__global__ __launch_bounds__(256) void moe_ln(float* __restrict__ out,
                                              const float* __restrict__ gamma,
                                              const float* __restrict__ beta) {
    const int t = blockIdx.x;
    float* row = out + (size_t)t * H_;
    const int tid = threadIdx.x;

    float v0 = row[tid];
    float v1 = row[tid + 256];

    __shared__ float s1[256];
    __shared__ float s2[256];
    s1[tid] = v0 + v1;
    s2[tid] = v0 * v0 + v1 * v1;
    __syncthreads();
    #pragma unroll
    for (int s = 128; s > 0; s >>= 1) {
        if (tid < s) { s1[tid] += s1[tid + s]; s2[tid] += s2[tid + s]; }
        __syncthreads();
    }
    const float mu   = s1[0] * (1.0f / H_);
    const float var  = s2[0] * (1.0f / H_) - mu * mu;
    const float rstd = rsqrtf(var + 1e-12f);

    row[tid]       = (v0 - mu) * rstd * gamma[tid]       + beta[tid];
    row[tid + 256] = (v1 - mu) * rstd * gamma[tid + 256] + beta[tid + 256];
}

// ---------------------------------------------------------------------------
// launch
// ---------------------------------------------------------------------------
extern "C" void kernel_launch(void* const* d_in, const int* in_sizes, int n_in,
                              void* d_out, int out_size, void* d_ws, size_t ws_size,
                              hipStream_t stream) {
    const float* x      = (const float*)d_in[0];   // [N_, H_]
    const int*   assign = (const int*)  d_in[1];   // [N_]
    const float* W1     = (const float*)d_in[2];   // [E_, H_, I_]
    const float* b1     = (const float*)d_in[3];   // [E_, I_]
    const float* W2     = (const float*)d_in[4];   // [E_, I_, H_]
    const float* b2     = (const float*)d_in[5];   // [E_, H_]
    const float* gamma  = (const float*)d_in[6];   // [H_]
    const float* beta   = (const float*)d_in[7];   // [H_]
    float*       out    = (float*)d_out;           // [N_, H_]

    // workspace layout
    int*   counters = (int*)d_ws;                                   // E_ ints
    int*   tos      = (int*)((char*)d_ws + 256);                    // E_*C_ ints
    float* hbuf     = (float*)((char*)d_ws + 256 + (size_t)E_ * C_ * sizeof(int));

    moe_init <<<(E_ * C_ + 255) / 256, 256, 0, stream>>>(counters, tos);
    moe_route<<<N_ / 256,            256, 0, stream>>>(assign, counters, tos);
    moe_seed_out<<<2048,             256, 0, stream>>>(x, out);
    moe_gemm1<<<dim3(I_ / TN, C_ / TM, E_), 256, 0, stream>>>(x, W1, b1, tos, counters, hbuf);
    moe_gemm2<<<dim3(H_ / TN, C_ / TM, E_), 256, 0, stream>>>(hbuf, W2, b2, x, tos, counters, out);
    moe_ln   <<<N_,                  256, 0, stream>>>(out, gamma, beta);
}